// Z_encoder_10625749090479
// MI455X (gfx1250) — compile-verified
//
#include <hip/hip_runtime.h>
#include <math.h>

#define NN 1024
#define DD 64
#define S_STEPS 2
#define NEG_BIG (-9.0e15f)

typedef float v2f __attribute__((ext_vector_type(2)));
typedef float v8f __attribute__((ext_vector_type(8)));

// ---------------------------------------------------------------- init
__global__ void init_k(const float* __restrict__ a_in, float* __restrict__ a_cur,
                       float* __restrict__ gsum) {
  int o = threadIdx.x;
  a_cur[o] = a_in[o];
  gsum[o]  = 0.f;
}

// ------------------------------------------------- z = emb@W, zs, zd
__global__ void prep_z_k(const float* __restrict__ emb, const float* __restrict__ W,
                         const float* __restrict__ a_cur,
                         float* __restrict__ z, float* __restrict__ zs,
                         float* __restrict__ zd) {
  __shared__ float er[DD];
  __shared__ float red[DD];
  int row = blockIdx.x;
  int d = threadIdx.x;
  er[d] = emb[(size_t)row * DD + d];
  __syncthreads();
  float acc = 0.f;
#pragma unroll 8
  for (int k = 0; k < DD; ++k) acc = fmaf(er[k], W[k * DD + d], acc);
  z[(size_t)row * DD + d] = acc;
  red[d] = acc * a_cur[d];          // a_src = a[:D]
  __syncthreads();
  if (d == 0) { float s = 0.f; for (int k = 0; k < DD; ++k) s += red[k]; zs[row] = s; }
  __syncthreads();
  red[d] = acc * a_cur[DD + d];     // a_dst = a[D:]
  __syncthreads();
  if (d == 0) { float s = 0.f; for (int k = 0; k < DD; ++k) s += red[k]; zd[row] = s; }
}

// ----------------- per-step trig fold: a_s*sin(m)+a_c*cos(m) = R*sin(m+phi)
__global__ void trig_k(const float* __restrict__ a_cur, float* __restrict__ trigR,
                       float* __restrict__ trigP) {
  int d = threadIdx.x;            // 64 threads
  float s = a_cur[2 * d];
  float c = a_cur[2 * d + 1];
  trigR[d] = sqrtf(s * s + c * c);
  trigP[d] = atan2f(c, s);
}

// ----------------- masked attention scores e[i][j] (dominant: v_sin_f32)
__global__ void __launch_bounds__(256) escore_k(
    int step, const int* __restrict__ adj, const float* __restrict__ ts,
    const int* __restrict__ tspan, const float* __restrict__ bfreq,
    const float* __restrict__ trigR, const float* __restrict__ trigP,
    const float* __restrict__ zs, const float* __restrict__ zd,
    float* __restrict__ E)
{
  __shared__ float bf[64], R[64], P[64];
  int tid = threadIdx.x;
  if (tid < 64) { bf[tid] = bfreq[tid]; R[tid] = trigR[tid]; P[tid] = trigP[tid]; }
  __syncthreads();
  size_t idx = (size_t)blockIdx.x * 256 + tid;
  int i = (int)(idx >> 10);
  int j = (int)(idx & 1023);
  float time_end = (float)((step + 1) * tspan[0]);
  float dt = time_end - ts[idx];
  float te = 0.f;
#pragma unroll 8
  for (int d = 0; d < 64; ++d)
    te = fmaf(R[d], __sinf(fmaf(dt, bf[d], P[d])), te);
  float e = zs[i] + zd[j] + te;
  e = (e >= 0.f) ? e : 0.05f * e;                 // leaky_relu(0.05)
  E[idx] = (adj[idx] > 0) ? e : NEG_BIG;
}

// --------------------------- softmax + att@z (WMMA f32) + residual + LN
__global__ void __launch_bounds__(256) attn_k(
    const float* __restrict__ E, const float* __restrict__ z,
    const float* __restrict__ gamma, const float* __restrict__ beta,
    float* __restrict__ outp)
{
  __shared__ float accs[16][64];
  __shared__ float red[16][16];
  __shared__ float red2[16][16];
  __shared__ float rowmax[16], rowsum[16];

  int tid  = threadIdx.x;
  int lane = tid & 31;
  int wid  = tid >> 5;
  int r0   = blockIdx.x * 16;
  int row  = tid >> 4;           // 0..15 local row
  int cseg = tid & 15;           // 16 threads cooperate per row
  int gr   = r0 + row;

  for (int i = tid; i < 16 * 64; i += 256) ((float*)accs)[i] = 0.f;

  // phase 1: row max over 1024 masked scores
  float lmax = NEG_BIG;
  for (int k = 0; k < 64; ++k)
    lmax = fmaxf(lmax, E[(size_t)gr * NN + cseg + 16 * k]);
  red[row][cseg] = lmax;
  __syncthreads();
  if (tid < 16) {
    float m = red[tid][0];
    for (int k = 1; k < 16; ++k) m = fmaxf(m, red[tid][k]);
    rowmax[tid] = m;
  }
  __syncthreads();

  // phase 2: row sum of exp
  float mx = rowmax[row];
  float lsum = 0.f;
  for (int k = 0; k < 64; ++k)
    lsum += __expf(E[(size_t)gr * NN + cseg + 16 * k] - mx);
  red[row][cseg] = lsum;
  __syncthreads();
  if (tid < 16) {
    float s = 0.f;
    for (int k = 0; k < 16; ++k) s += red[tid][k];
    rowsum[tid] = s;
  }
  __syncthreads();

  // phase 3: P @ z via V_WMMA_F32_16X16X4_F32 (p computed on the fly)
  v8f cacc[4];
#pragma unroll
  for (int c = 0; c < 4; ++c)
#pragma unroll
    for (int r = 0; r < 8; ++r) cacc[c][r] = 0.f;

  int arow = lane & 15;
  int koff = (lane < 16) ? 0 : 2;        // ISA: lanes 16-31 hold K=2,3
  float amax = rowmax[arow];

  for (int kc = wid; kc < 256; kc += 8) {   // 256 K-chunks of 4, 8 waves
    int k0 = kc * 4;
    v2f av;
    av.x = __expf(E[(size_t)(r0 + arow) * NN + k0 + koff]     - amax);
    av.y = __expf(E[(size_t)(r0 + arow) * NN + k0 + koff + 1] - amax);
    int gj = k0 + koff;
#pragma unroll
    for (int c = 0; c < 4; ++c) {
      v2f bv;
      int col = c * 16 + arow;
      bv.x = z[(size_t)(gj    ) * DD + col];
      bv.y = z[(size_t)(gj + 1) * DD + col];
      cacc[c] = __builtin_amdgcn_wmma_f32_16x16x4_f32(
          false, av, false, bv, (short)0, cacc[c], false, false);
    }
  }

  // combine per-wave partials into LDS (ds_add_f32)
#pragma unroll
  for (int c = 0; c < 4; ++c)
#pragma unroll
    for (int r = 0; r < 8; ++r) {
      int rr = r + ((lane < 16) ? 0 : 8);
      atomicAdd(&accs[rr][c * 16 + arow], cacc[c][r]);
    }
  __syncthreads();

  // epilogue: temp = att@z + z, LayerNorm (parallel, 4 cols/thread)
  int c0 = cseg * 4;
  float inv = 1.0f / rowsum[row];
  float tv[4];
  float pmu = 0.f, psq = 0.f;
#pragma unroll
  for (int q = 0; q < 4; ++q) {
    float t = accs[row][c0 + q] * inv + z[(size_t)gr * DD + c0 + q];
    tv[q] = t; pmu += t; psq += t * t;
  }
  red[row][cseg]  = pmu;
  red2[row][cseg] = psq;
  __syncthreads();
  if (tid < 16) {
    float s = 0.f, q = 0.f;
    for (int k = 0; k < 16; ++k) { s += red[tid][k]; q += red2[tid][k]; }
    float mu  = s * (1.0f / 64.0f);
    float var = q * (1.0f / 64.0f) - mu * mu;
    rowmax[tid] = mu;                        // reuse as mu
    rowsum[tid] = rsqrtf(var + 1e-6f);       // reuse as inv-std
  }
  __syncthreads();
  float mu = rowmax[row], rs = rowsum[row];
#pragma unroll
  for (int q = 0; q < 4; ++q)
    outp[(size_t)gr * DD + c0 + q] =
        gamma[c0 + q] * (tv[q] - mu) * rs + beta[c0 + q];
}

// ------------------------------ GRU: hid@Wh + bh (hid identical per row)
__global__ void gru_const_k(const float* __restrict__ a_cur, const float* __restrict__ wh,
                            const float* __restrict__ bh, float* __restrict__ cbuf) {
  __shared__ float ah[128];
  int o = threadIdx.x;
  ah[o] = a_cur[o];
  __syncthreads();
  for (int k = 0; k < 3; ++k) {
    float s = bh[k * 128 + o];
#pragma unroll 4
    for (int h = 0; h < 128; ++h) s = fmaf(ah[h], wh[k * 128 * 128 + h * 128 + o], s);
    cbuf[k * 128 + o] = s;
  }
}

// ------------------------------ GRU per-row + mean accumulation
__global__ void gru_row_k(const float* __restrict__ emb, const float* __restrict__ wi,
                          const float* __restrict__ bi, const float* __restrict__ cbuf,
                          const float* __restrict__ a_cur, float* __restrict__ gsum) {
  __shared__ float x[64];
  int o = threadIdx.x;
  int row = blockIdx.x;
  if (o < 64) x[o] = emb[(size_t)row * 64 + o];
  __syncthreads();
  float xi[3];
  for (int k = 0; k < 3; ++k) {
    float s = bi[k * 128 + o];
#pragma unroll 4
    for (int d = 0; d < 64; ++d) s = fmaf(x[d], wi[k * 64 * 128 + d * 128 + o], s);
    xi[k] = s;
  }
  float r  = 1.f / (1.f + __expf(-(xi[0] + cbuf[o])));
  float zg = 1.f / (1.f + __expf(-(xi[1] + cbuf[128 + o])));
  float n  = tanhf(xi[2] + r * cbuf[256 + o]);
  float h  = (1.f - zg) * n + zg * a_cur[o];
  atomicAdd(&gsum[o], h);
}

// ------------------------------ finalize: a <- mean, reset, emit tail
__global__ void fin_k(float* __restrict__ gsum, float* __restrict__ a_cur,
                      float* __restrict__ out_tail) {
  int o = threadIdx.x;
  float v = gsum[o] * (1.0f / (float)NN);
  a_cur[o]    = v;
  gsum[o]     = 0.f;
  out_tail[o] = v;
}

extern "C" void kernel_launch(void* const* d_in, const int* in_sizes, int n_in,
                              void* d_out, int out_size, void* d_ws, size_t ws_size,
                              hipStream_t stream) {
  const int*   adj   = (const int*)  d_in[0];
  const float* feat  = (const float*)d_in[1];
  const float* ts    = (const float*)d_in[2];
  const int*   tspan = (const int*)  d_in[3];
  const float* W     = (const float*)d_in[4];
  const float* a_in  = (const float*)d_in[5];
  const float* bfreq = (const float*)d_in[6];
  const float* gamma = (const float*)d_in[7];
  const float* beta  = (const float*)d_in[8];
  const float* wi    = (const float*)d_in[9];
  const float* wh    = (const float*)d_in[10];
  const float* bi    = (const float*)d_in[11];
  const float* bh    = (const float*)d_in[12];
  float* out = (float*)d_out;
  float* ws  = (float*)d_ws;

  float* z     = ws;                // N*D   = 65536
  float* zsv   = ws + 65536;        // N
  float* zdv   = ws + 66560;        // N
  float* a_cur = ws + 67584;        // 128
  float* cbuf  = ws + 67712;        // 3*128
  float* gsum  = ws + 68096;        // 128
  float* trigR = ws + 68224;        // 64
  float* trigP = ws + 68288;        // 64
  float* E     = ws + 68352;        // N*N = 1048576

  init_k<<<1, 128, 0, stream>>>(a_in, a_cur, gsum);
  for (int s = 0; s < S_STEPS; ++s) {
    const float* emb = (s == 0) ? feat : (out + (size_t)(s - 1) * NN * DD);
    prep_z_k<<<NN, 64, 0, stream>>>(emb, W, a_cur, z, zsv, zdv);
    trig_k<<<1, 64, 0, stream>>>(a_cur, trigR, trigP);
    escore_k<<<(NN * NN) / 256, 256, 0, stream>>>(
        s, adj + (size_t)s * NN * NN, ts + (size_t)s * NN * NN, tspan,
        bfreq, trigR, trigP, zsv, zdv, E);
    attn_k<<<NN / 16, 256, 0, stream>>>(E, z, gamma, beta,
                                        out + (size_t)s * NN * DD);
    gru_const_k<<<1, 128, 0, stream>>>(a_cur, wh, bh, cbuf);
    gru_row_k<<<NN, 128, 0, stream>>>(out + (size_t)s * NN * DD, wi, bi, cbuf,
                                      a_cur, gsum);
    fin_k<<<1, 128, 0, stream>>>(gsum, a_cur, out + (size_t)S_STEPS * NN * DD);
  }
}